// SingleNodeReadout_79937931313654
// MI455X (gfx1250) — compile-verified
//
#include <hip/hip_runtime.h>
#include <math.h>

// ---------------- problem constants (from reference) ----------------
#define B_      4
#define T_      12
#define P_      2000
#define N_      20000
#define FPF     16          // patch feature width
#define FNF     8           // node feature width
#define H_      12
#define E_      160000
#define IN_DIM  288         // T*(FPF+FNF)
#define OUT_DIM 24          // H*2
#define ROWS    (B_ * N_)   // 80000
#define NODES_OFF 96        // T_*FNF : nodes_flat cols [0,96), patch cols [96,288)
#define LDS_STRIDE 292      // 288 + 4 floats pad (divisible by 4 -> float4-able)
#define MROWS   64          // rows per block (4 x 16 M-tiles)

typedef float v2f __attribute__((ext_vector_type(2)));
typedef float v8f __attribute__((ext_vector_type(8)));

// ---------------- tiny helpers ----------------
__global__ void fill_zero_kernel(float* __restrict__ p, int n) {
    int i = blockIdx.x * blockDim.x + threadIdx.x;
    if (i < n) p[i] = 0.0f;
}

// counts[n] = number of edges mapping to node n (float histogram)
__global__ void count_kernel(const int* __restrict__ mapper, float* __restrict__ counts) {
    int e = blockIdx.x * blockDim.x + threadIdx.x;
    if (e < E_) atomicAdd(&counts[mapper[e]], 1.0f);
}

// For each edge e and feature f: for all (b,t),
//   mlp_in[(b*N+n)*288 + 96 + t*16 + f] += patch_x[b,t,p,f]
__global__ void scatter_kernel(const float* __restrict__ patch_x,
                               const int* __restrict__ sb,
                               const int* __restrict__ snm,
                               float* __restrict__ mlp_in) {
    int idx = blockIdx.x * blockDim.x + threadIdx.x;
    if (idx >= E_ * FPF) return;
    int e = idx >> 4;            // / FPF
    int f = idx & (FPF - 1);
    int p = sb[e];
    int n = snm[e];
    for (int b = 0; b < B_; ++b) {
        float*       dst = mlp_in + (size_t)(b * N_ + n) * IN_DIM + NODES_OFF + f;
        const float* src = patch_x + ((size_t)(b * T_) * P_ + p) * FPF + f;
        for (int t = 0; t < T_; ++t) {
            atomicAdd(dst + t * FPF, src[(size_t)t * P_ * FPF]);
        }
    }
}

// Normalize patch section by 1/count and fill nodes section from nodes_x.
__global__ void norm_fill_kernel(const float* __restrict__ nodes_x,
                                 const float* __restrict__ counts,
                                 float* __restrict__ mlp_in) {
    long long idx = (long long)blockIdx.x * blockDim.x + threadIdx.x;
    if (idx >= (long long)ROWS * IN_DIM) return;
    int row = (int)(idx / IN_DIM);
    int col = (int)(idx - (long long)row * IN_DIM);
    int b = row / N_;
    int n = row - b * N_;
    if (col < NODES_OFF) {
        int t = col / FNF, f = col - t * FNF;
        mlp_in[idx] = nodes_x[(((size_t)(b * T_ + t)) * N_ + n) * FNF + f];
    } else {
        float cnt = counts[n];
        float scale = (cnt > 0.0f) ? (1.0f / cnt) : 0.0f;
        mlp_in[idx] *= scale;
    }
}

// ---------------- fused MLP: relu(X*W1+b1)*W2+b2, WMMA fp32 path ----------------
// Block: 128 threads = 4 wave32s, handles 64 rows of X (4 M-tiles of 16).
// GEMM1: waves split the 18 column tiles; each B-fragment (W1 column pair)
// feeds 4 WMMAs (one per M-tile) from registers -> 4x less W1/L2 traffic.
// GEMM2: wave w owns M-tile w (16 rows x 24 cols).
// WMMA V_WMMA_F32_16X16X4_F32 fragment layouts (ISA 7.12.2):
//   A 16x4 : lane&15 is the row; VGPR v holds K = v + 2*(lane>>4)
//   B 4x16 : lane&15 is the col; VGPR v holds K = v + 2*(lane>>4)
//   C 16x16: VGPR r holds (M = r + 8*(lane>>4), N = (lane&15) + col0)
__global__ __launch_bounds__(128) void mlp_wmma_kernel(
        const float* __restrict__ mlp_in,
        const float* __restrict__ W1, const float* __restrict__ b1,
        const float* __restrict__ W2, const float* __restrict__ b2,
        float* __restrict__ out) {
    __shared__ float A_lds[MROWS * LDS_STRIDE];   // 64 x 288 (padded)
    __shared__ float H_lds[MROWS * LDS_STRIDE];   // relu(X*W1+b1) tile

    const int tid  = threadIdx.x;
    const int lane = tid & 31;
    const int wave = tid >> 5;
    const int half = lane >> 4;     // 0: lanes 0-15, 1: lanes 16-31
    const int cN   = lane & 15;     // B/C/D col within tile; A row within M-tile
    const int row0 = blockIdx.x * MROWS;

    // Stage the 64x288 A-tile into LDS with float4 (global_load_b128).
    {
        const float4* __restrict__ g4 = (const float4*)mlp_in + (size_t)row0 * (IN_DIM / 4);
        float4* __restrict__ l4 = (float4*)A_lds;
        const int RQ = IN_DIM / 4;            // 72 float4 per row
        for (int i = tid; i < MROWS * RQ; i += 128) {
            int mr = i / RQ, mc = i - mr * RQ;
            l4[mr * (LDS_STRIDE / 4) + mc] = g4[i];
        }
    }
    __syncthreads();

    // ---- GEMM1: h = relu(A * W1 + b1); 18 column tiles over 4 waves ----
    for (int tn = wave; tn < IN_DIM / 16; tn += 4) {
        const int col0 = tn * 16;
        const float* __restrict__ w1p = W1 + col0 + cN;
        v8f acc0 = {}, acc1 = {}, acc2 = {}, acc3 = {};
        for (int k = 0; k < IN_DIM; k += 4) {
            const int ka = k + 2 * half;
            v2f bf;
            bf.x = w1p[(size_t)ka * IN_DIM];
            bf.y = w1p[(size_t)(ka + 1) * IN_DIM];
            v2f a0, a1, a2, a3;
            a0.x = A_lds[(0 * 16 + cN) * LDS_STRIDE + ka];
            a0.y = A_lds[(0 * 16 + cN) * LDS_STRIDE + ka + 1];
            a1.x = A_lds[(1 * 16 + cN) * LDS_STRIDE + ka];
            a1.y = A_lds[(1 * 16 + cN) * LDS_STRIDE + ka + 1];
            a2.x = A_lds[(2 * 16 + cN) * LDS_STRIDE + ka];
            a2.y = A_lds[(2 * 16 + cN) * LDS_STRIDE + ka + 1];
            a3.x = A_lds[(3 * 16 + cN) * LDS_STRIDE + ka];
            a3.y = A_lds[(3 * 16 + cN) * LDS_STRIDE + ka + 1];
            acc0 = __builtin_amdgcn_wmma_f32_16x16x4_f32(false, a0, false, bf, (short)0, acc0, false, false);
            acc1 = __builtin_amdgcn_wmma_f32_16x16x4_f32(false, a1, false, bf, (short)0, acc1, false, false);
            acc2 = __builtin_amdgcn_wmma_f32_16x16x4_f32(false, a2, false, bf, (short)0, acc2, false, false);
            acc3 = __builtin_amdgcn_wmma_f32_16x16x4_f32(false, a3, false, bf, (short)0, acc3, false, false);
        }
        const float bias = b1[col0 + cN];
#pragma unroll
        for (int rr = 0; rr < 8; ++rr) {
            const int m = rr + 8 * half;
            float v0 = acc0[rr] + bias;
            float v1 = acc1[rr] + bias;
            float v2 = acc2[rr] + bias;
            float v3 = acc3[rr] + bias;
            H_lds[(0 * 16 + m) * LDS_STRIDE + col0 + cN] = v0 > 0.0f ? v0 : 0.0f;
            H_lds[(1 * 16 + m) * LDS_STRIDE + col0 + cN] = v1 > 0.0f ? v1 : 0.0f;
            H_lds[(2 * 16 + m) * LDS_STRIDE + col0 + cN] = v2 > 0.0f ? v2 : 0.0f;
            H_lds[(3 * 16 + m) * LDS_STRIDE + col0 + cN] = v3 > 0.0f ? v3 : 0.0f;
        }
    }
    __syncthreads();

    // ---- GEMM2: out = h * W2 + b2 (each wave: one 16-row tile x 24 cols) ----
    const int mrow0 = wave * 16;
#pragma unroll
    for (int ct = 0; ct < 2; ++ct) {
        const int col0 = ct * 16;
        const int col  = col0 + cN;            // 0..31, valid if < 24
        const bool colOK = (col < OUT_DIM);
        v8f acc = {};
        for (int k = 0; k < IN_DIM; k += 4) {
            const int ka = k + 2 * half;
            v2f a, bf;
            a.x  = H_lds[(mrow0 + cN) * LDS_STRIDE + ka];
            a.y  = H_lds[(mrow0 + cN) * LDS_STRIDE + ka + 1];
            bf.x = colOK ? W2[(size_t)ka * OUT_DIM + col] : 0.0f;
            bf.y = colOK ? W2[(size_t)(ka + 1) * OUT_DIM + col] : 0.0f;
            acc = __builtin_amdgcn_wmma_f32_16x16x4_f32(false, a, false, bf, (short)0, acc, false, false);
        }
        if (colOK) {
            const float bias = b2[col];
            const int hh = col >> 1;
            const int parity = col & 1;
#pragma unroll
            for (int rr = 0; rr < 8; ++rr) {
                const int m    = rr + 8 * half;
                const int grow = row0 + mrow0 + m;
                const int bb   = grow / N_;
                const int nn   = grow - bb * N_;
                const float v  = acc[rr] + bias;
                const size_t base = ((size_t)(bb * H_ + hh)) * N_ + nn;
                if (parity == 0) {
                    out[base] = v;                                 // mu
                } else {
                    float sp = (v > 20.0f) ? v : log1pf(expf(v));  // softplus
                    out[base + (size_t)B_ * H_ * N_] = sp + 1e-6f; // sigma
                }
            }
        }
    }
}

// ---------------- launch ----------------
extern "C" void kernel_launch(void* const* d_in, const int* in_sizes, int n_in,
                              void* d_out, int out_size, void* d_ws, size_t ws_size,
                              hipStream_t stream) {
    const float* patch_x = (const float*)d_in[0];
    const float* nodes_x = (const float*)d_in[1];
    const int*   sb      = (const int*)d_in[2];
    const int*   snm     = (const int*)d_in[3];
    const float* W1      = (const float*)d_in[4];
    const float* b1      = (const float*)d_in[5];
    const float* W2      = (const float*)d_in[6];
    const float* b2      = (const float*)d_in[7];
    float*       out     = (float*)d_out;

    // Workspace layout: mlp_in (ROWS x 288 fp32) | counts (N fp32)
    float* mlp_in = (float*)d_ws;
    float* counts = mlp_in + (size_t)ROWS * IN_DIM;

    const int zn = ROWS * IN_DIM + N_;   // zero mlp_in + counts (contiguous)
    fill_zero_kernel<<<(zn + 255) / 256, 256, 0, stream>>>(mlp_in, zn);

    count_kernel<<<(E_ + 255) / 256, 256, 0, stream>>>(snm, counts);

    scatter_kernel<<<(E_ * FPF + 255) / 256, 256, 0, stream>>>(patch_x, sb, snm, mlp_in);

    const long long nf = (long long)ROWS * IN_DIM;
    norm_fill_kernel<<<(unsigned)((nf + 255) / 256), 256, 0, stream>>>(nodes_x, counts, mlp_in);

    mlp_wmma_kernel<<<ROWS / MROWS, 128, 0, stream>>>(mlp_in, W1, b1, W2, b2, out);
}